// BasketballGNN_46583215292447
// MI455X (gfx1250) — compile-verified
//
#include <hip/hip_runtime.h>
#include <cstddef>

typedef __attribute__((ext_vector_type(2))) float v2f;
typedef __attribute__((ext_vector_type(8))) float v8f;

#define GCN_N 50000
#define THREADS 256

// ---------------------------------------------------------------------------
// Degree / normalization:  deg[i] = indegree(i) + 1 (self loop);  dinv = rsqrt
// ---------------------------------------------------------------------------
__global__ void k_init_deg(float* __restrict__ deg, int n) {
    int i = blockIdx.x * blockDim.x + threadIdx.x;
    if (i < n) deg[i] = 1.0f;  // self-loop contribution
}

__global__ void k_accum_deg(const int* __restrict__ dst, float* __restrict__ deg, int e) {
    int i = blockIdx.x * blockDim.x + threadIdx.x;
    if (i < e) atomicAdd(&deg[dst[i]], 1.0f);
}

__global__ void k_dinv(float* __restrict__ deg, int n) {
    int i = blockIdx.x * blockDim.x + threadIdx.x;
    if (i < n) deg[i] = rsqrtf(fmaxf(deg[i], 1.0f));
}

__global__ void k_zero(float* __restrict__ p, size_t n) {
    size_t i = (size_t)blockIdx.x * blockDim.x + threadIdx.x;
    if (i < n) p[i] = 0.0f;
}

// ---------------------------------------------------------------------------
// Dense transform C[N x KOUT] = A[N x KIN] @ W[KIN x KOUT] via
// V_WMMA_F32_16X16X4_F32.  One wave32 owns a 16-row tile.
//
// W is staged in LDS *K-pair interleaved*:
//   sW[k2*KOUT + c] = { W[2*k2][c], W[2*k2+1][c] }   (v2f, 8B aligned)
// so each lane's B fragment (two consecutive K rows, same column) is a single
// ds_load_b64 straight into an even-aligned VGPR pair -> no v_mov shuffles.
//
// VGPR layouts (ISA 7.12.2):
//   A 16x4 f32 : lane m (0..15) = row m; VGPR0 = K {0 | 2}, VGPR1 = K {1 | 3}
//                (lanes 16..31 hold the K+2 columns)
//   B 4x16 f32 : VGPR0 = K-rows {0 | 2}, VGPR1 = {1 | 3}, N = lane & 15
//   C/D 16x16  : VGPR v = rows {v | v+8}, N = lane & 15
// ---------------------------------------------------------------------------
template <int KIN, int KOUT>
__global__ __launch_bounds__(THREADS)
void k_gemm_wmma(const float* __restrict__ A, const float* __restrict__ W,
                 float* __restrict__ C, int nrows) {
    __shared__ v2f sW[(KIN / 2) * KOUT];
    for (int i = threadIdx.x; i < (KIN / 2) * KOUT; i += blockDim.x) {
        const int k2 = i / KOUT;
        const int c  = i % KOUT;
        v2f w;
        w.x = W[(2 * k2 + 0) * KOUT + c];
        w.y = W[(2 * k2 + 1) * KOUT + c];
        sW[i] = w;
    }
    __syncthreads();

    const int wave = threadIdx.x >> 5;            // 8 waves / block
    const int lane = threadIdx.x & 31;
    const int tile = blockIdx.x * 8 + wave;       // 16-row tile index
    if (tile * 16 >= nrows) return;               // wave-uniform: EXEC stays full

    constexpr int NT = KOUT / 16;                 // output column tiles
    v8f acc[NT] = {};

    const int m    = lane & 15;                   // row (A) / column (B,C)
    const int khi  = lane >> 4;                   // 0: K{0,1}, 1: K{2,3}
    const float* Arow = A + (size_t)(tile * 16 + m) * KIN;
    const v2f* sWl = sW + khi * KOUT + m;         // per-lane base into pair table

    for (int kk = 0; kk < KIN / 4; ++kk) {
        // A fragment: one global_load_b64 (even K offset -> 8B aligned)
        const v2f a = *(const v2f*)(Arow + kk * 4 + khi * 2);
        const v2f* sWk = sWl + (kk * 2) * KOUT;   // pair row = kk*2 + khi
#pragma unroll
        for (int t = 0; t < NT; ++t) {
            const v2f b = sWk[t * 16];            // single ds_load_b64
            acc[t] = __builtin_amdgcn_wmma_f32_16x16x4_f32(
                /*neg_a=*/false, a, /*neg_b=*/false, b,
                /*c_mod=*/(short)0, acc[t], /*reuse_a=*/false, /*reuse_b=*/false);
        }
    }

#pragma unroll
    for (int t = 0; t < NT; ++t) {
#pragma unroll
        for (int v = 0; v < 8; ++v) {
            const int r = tile * 16 + khi * 8 + v;
            C[(size_t)r * KOUT + t * 16 + m] = acc[t][v];
        }
    }
}

// ---------------------------------------------------------------------------
// Edge aggregation: out[dst] += H[src] * dinv[src]*dinv[dst]
// thread = (edge, 4-feature chunk); 16 (F=64) / 8 (F=32) lanes cover one row,
// so source-row reads are coalesced 256B/128B bursts and atomics hit
// consecutive addresses (L2-resident given 192MB L2).
// ---------------------------------------------------------------------------
template <int F>
__global__ __launch_bounds__(THREADS)
void k_scatter(const int* __restrict__ src, const int* __restrict__ dst,
               const float* __restrict__ dinv, const float* __restrict__ H,
               float* __restrict__ out, int nedges) {
    constexpr int C = F / 4;
    const long long tid = (long long)blockIdx.x * blockDim.x + threadIdx.x;
    const int e = (int)(tid / C);
    const int c = (int)(tid % C);
    if (e >= nedges) return;

    const int s = src[e];
    const int d = dst[e];
    const float nrm = dinv[s] * dinv[d];
    const float4 h = *(const float4*)(H + (size_t)s * F + c * 4);
    float* o = out + (size_t)d * F + c * 4;
    atomicAdd(o + 0, h.x * nrm);
    atomicAdd(o + 1, h.y * nrm);
    atomicAdd(o + 2, h.z * nrm);
    atomicAdd(o + 3, h.w * nrm);
}

// ---------------------------------------------------------------------------
// Finalize: out += self-loop message + bias, optional ReLU
// ---------------------------------------------------------------------------
template <int F, bool RELU>
__global__ __launch_bounds__(THREADS)
void k_finalize(const float* __restrict__ H, const float* __restrict__ dinv,
                const float* __restrict__ bias, float* __restrict__ out, int n) {
    const long long tid = (long long)blockIdx.x * blockDim.x + threadIdx.x;
    const int i = (int)(tid / F);
    const int f = (int)(tid % F);
    if (i >= n) return;
    const float di = dinv[i];
    float v = out[(size_t)i * F + f] + H[(size_t)i * F + f] * di * di + bias[f];
    out[(size_t)i * F + f] = RELU ? fmaxf(v, 0.0f) : v;
}

// ---------------------------------------------------------------------------
extern "C" void kernel_launch(void* const* d_in, const int* in_sizes, int n_in,
                              void* d_out, int out_size, void* d_ws, size_t ws_size,
                              hipStream_t stream) {
    const float* x  = (const float*)d_in[0];   // [N,128]
    const int*   ei = (const int*)d_in[1];     // [2,E]
    const float* W1 = (const float*)d_in[2];   // [128,64]
    const float* b1 = (const float*)d_in[3];
    const float* W2 = (const float*)d_in[4];   // [64,64]
    const float* b2 = (const float*)d_in[5];
    const float* W3 = (const float*)d_in[6];   // [64,32]
    const float* b3 = (const float*)d_in[7];
    float* out = (float*)d_out;                // [N,32]

    const int N = GCN_N;
    const int E = in_sizes[1] / 2;
    const int* src = ei;
    const int* dst = ei + E;

    // Workspace: dinv[N] | bufA[N*64] | bufB[N*64]  (~26 MB)
    float* dinv = (float*)d_ws;
    float* bufA = dinv + ((N + 255) & ~255);
    float* bufB = bufA + (size_t)N * 64;

    const int tiles  = N / 16;                     // 3125 (exact)
    const int gemmBl = (tiles + 7) / 8;

    auto blocks = [](long long n) { return (unsigned)((n + THREADS - 1) / THREADS); };

    // --- normalization ---
    k_init_deg<<<blocks(N), THREADS, 0, stream>>>(dinv, N);
    k_accum_deg<<<blocks(E), THREADS, 0, stream>>>(dst, dinv, E);
    k_dinv<<<blocks(N), THREADS, 0, stream>>>(dinv, N);

    // --- layer 1: H1 = relu(agg(x @ W1) + b1) ---
    k_gemm_wmma<128, 64><<<gemmBl, THREADS, 0, stream>>>(x, W1, bufA, N);
    k_zero<<<blocks((long long)N * 64), THREADS, 0, stream>>>(bufB, (size_t)N * 64);
    k_scatter<64><<<blocks((long long)E * 16), THREADS, 0, stream>>>(src, dst, dinv, bufA, bufB, E);
    k_finalize<64, true><<<blocks((long long)N * 64), THREADS, 0, stream>>>(bufA, dinv, b1, bufB, N);

    // --- layer 2: H2 = relu(agg(H1 @ W2) + b2) ---
    k_gemm_wmma<64, 64><<<gemmBl, THREADS, 0, stream>>>(bufB, W2, bufA, N);
    k_zero<<<blocks((long long)N * 64), THREADS, 0, stream>>>(bufB, (size_t)N * 64);
    k_scatter<64><<<blocks((long long)E * 16), THREADS, 0, stream>>>(src, dst, dinv, bufA, bufB, E);
    k_finalize<64, true><<<blocks((long long)N * 64), THREADS, 0, stream>>>(bufA, dinv, b2, bufB, N);

    // --- layer 3: out = agg(H2 @ W3) + b3 ---
    k_gemm_wmma<64, 32><<<gemmBl, THREADS, 0, stream>>>(bufB, W3, bufA, N);
    k_zero<<<blocks((long long)N * 32), THREADS, 0, stream>>>(out, (size_t)N * 32);
    k_scatter<32><<<blocks((long long)E * 8), THREADS, 0, stream>>>(src, dst, dinv, bufA, out, E);
    k_finalize<32, false><<<blocks((long long)N * 32), THREADS, 0, stream>>>(bufA, dinv, b3, out, N);
}